// QueryAndGroup_20882130993668
// MI455X (gfx1250) — compile-verified
//
#include <hip/hip_runtime.h>

// Problem constants (fixed by setup_inputs)
#define BB 4
#define NN 16384
#define MM 4096
#define CC 64
#define NS 32
#define CH (CC + 3)           // 67 output channels in new_features
#define WSPLIT 4              // waves per block cooperating on one 16-query tile
#define RANGE (NN / WSPLIT)   // points scanned per wave

typedef float v2f __attribute__((ext_vector_type(2)));
typedef float v8f __attribute__((ext_vector_type(8)));

#define EMPTYK 0xFFFFFFFFu    // empty-slot sentinel: above every real key

// Scores are true d^2 >= 0 (C seeded with |q|^2), so IEEE bits are already
// unsigned-order-monotonic. One v_max_i32 clamps the rare epsilon-negative
// rounding case (catastrophic cancellation) to key 0 = "nearest", where such
// points belong anyway.
__device__ __forceinline__ unsigned f2key(float x) {
  int u = __builtin_bit_cast(int, x);
  return (unsigned)(u > 0 ? u : 0);
}

// ---- DPP row-rotate reductions: rows are 16 lanes on gfx1250, exactly our
// half-wave query groups. ROW_ROR:n = 0x120+n. Rotate-reduce over {1,2,4,8}
// leaves every lane of the 16-lane row with the row-wide max/min; these fuse
// into single v_max_u32_dpp / v_min_u32_dpp instructions.
template <int CTRL>
__device__ __forceinline__ unsigned dpp_mov_u32(unsigned x) {
  return (unsigned)__builtin_amdgcn_update_dpp((int)x, (int)x, CTRL, 0xF, 0xF, true);
}
__device__ __forceinline__ unsigned row_max16_u32(unsigned v) {
  unsigned t;
  t = dpp_mov_u32<0x121>(v); v = (v > t) ? v : t;
  t = dpp_mov_u32<0x122>(v); v = (v > t) ? v : t;
  t = dpp_mov_u32<0x124>(v); v = (v > t) ? v : t;
  t = dpp_mov_u32<0x128>(v); v = (v > t) ? v : t;
  return v;
}
__device__ __forceinline__ unsigned row_min16_u32(unsigned v) {
  unsigned t;
  t = dpp_mov_u32<0x121>(v); v = (v < t) ? v : t;
  t = dpp_mov_u32<0x122>(v); v = (v < t) ? v : t;
  t = dpp_mov_u32<0x124>(v); v = (v < t) ? v : t;
  t = dpp_mov_u32<0x128>(v); v = (v < t) ? v : t;
  return v;
}

// Streaming top-32 insert in the integer-key domain. The 32-entry set for one
// query lives in a 16-lane half-wave: 2 (key,id) slots per lane. cm = current
// worst (max) key of the set. Both halves run in lockstep; all 32 lanes stay
// active. Leader broadcast uses v_readlane with wave-uniform lane indices
// derived from the ballot mask -> zero LDS traffic in the hot loop.
// DERIVE_P: scan phase, p == pbase + (lane&15), leader's id is arithmetic.
template <bool DERIVE_P>
__device__ __forceinline__ void topk_insert(unsigned k, int p, int pbase,
                                            unsigned& b0, unsigned& b1,
                                            int& i0, int& i1,
                                            unsigned& cm, int lane, int h) {
  bool pend = (k < cm);
  unsigned mask;
  while ((mask = (unsigned)__ballot(pend)) != 0u) {
    const unsigned m0 = mask & 0xFFFFu;
    const unsigned m1 = mask >> 16;
    const int lead0 = m0 ? __builtin_ctz(m0) : 0;          // wave-uniform
    const int lead1 = m1 ? (16 + __builtin_ctz(m1)) : 16;  // wave-uniform
    const unsigned kd0 = (unsigned)__builtin_amdgcn_readlane((int)k, lead0);
    const unsigned kd1 = (unsigned)__builtin_amdgcn_readlane((int)k, lead1);
    const unsigned kd = (h == 0) ? kd0 : kd1;
    const int lead = (h == 0) ? lead0 : lead1;
    int sp;
    if (DERIVE_P) {
      sp = pbase + (lead & 15);
    } else {
      const int sp0 = __builtin_amdgcn_readlane(p, lead0);
      const int sp1 = __builtin_amdgcn_readlane(p, lead1);
      sp = (h == 0) ? sp0 : sp1;
    }
    const unsigned mym = (h == 0) ? m0 : m1;
    const bool have = (mym != 0u);
    // locate the worst entry (key == cm by construction) in this half
    const unsigned mymax = (b0 > b1) ? b0 : b1;
    const bool ocand = have && (mymax == cm);
    const unsigned om = ((unsigned)__ballot(ocand) >> (h * 16)) & 0xFFFFu;
    const int owner = h * 16 + __builtin_ctz(om | 0x10000u);
    if (have && lane == owner) {
      if (b0 >= b1) { b0 = kd; i0 = sp; }
      else          { b1 = kd; i1 = sp; }
    }
    cm = row_max16_u32((b0 > b1) ? b0 : b1);   // pure-VALU 16-lane reduction
    if (have && lane == lead) pend = false;
    pend = pend && (k < cm);
  }
}

// Phase 1: fused WMMA distance + in-register top-32 selection.
// Block = 128 threads = 4 waves; each block owns 16 queries of one batch.
// One V_WMMA_F32_16X16X4_F32 per 16-point chunk computes the full
// d^2 = |q|^2 - 2 q.p + |p|^2 for a 16x16 (query x point) tile:
//   A = [-2qx,-2qy,-2qz,1], B = [px,py,pz,|p|^2], C = |q|^2 (per-row).
// Point loads are software-pipelined one chunk ahead to hide load-to-use
// latency behind the WMMA + selection work.
__global__ __launch_bounds__(WSPLIT * 32) void
knn_kernel(const float* __restrict__ xyz, const float* __restrict__ new_xyz,
           int* __restrict__ idxbuf) {
  const int w    = threadIdx.x >> 5;
  const int lane = threadIdx.x & 31;
  const int h    = lane >> 4;     // half-wave (= DPP row) id
  const int t    = lane & 15;     // id within half
  const int b     = blockIdx.x / (MM / 16);
  const int mtile = blockIdx.x % (MM / 16);

  // A operand: lanes 0-15 hold K=0/K=1, lanes 16-31 hold K=2/K=3.
  const int mq_t = mtile * 16 + t;
  const float* qp = new_xyz + ((size_t)b * MM + mq_t) * 3;
  const float qx = qp[0], qy = qp[1], qz = qp[2];
  v2f av;
  av.x = (h == 0) ? (-2.0f * qx) : (-2.0f * qz);
  av.y = (h == 0) ? (-2.0f * qy) : 1.0f;

  // C operand: acc[j] must start at |q_{j+8h}|^2 (C layout: VGPR j, lanes
  // 0-15 -> query j, lanes 16-31 -> query j+8). One-time readlane gather.
  const float qn2 = qx * qx + qy * qy + qz * qz;  // lane t holds query t
  v8f cinit;
#pragma unroll
  for (int j = 0; j < 8; ++j) {
    const float lo = __builtin_bit_cast(float,
        __builtin_amdgcn_readlane(__builtin_bit_cast(int, qn2), j));
    const float hi = __builtin_bit_cast(float,
        __builtin_amdgcn_readlane(__builtin_bit_cast(int, qn2), j + 8));
    cinit[j] = (h == 0) ? lo : hi;
  }

  // per-lane top-32 state for 8 queries (query = mtile*16 + j + 8*h)
  unsigned bd0[8], bd1[8], cmax[8];
  int      bi0[8], bi1[8];
#pragma unroll
  for (int j = 0; j < 8; ++j) {
    bd0[j] = EMPTYK; bd1[j] = EMPTYK; cmax[j] = EMPTYK;
    bi0[j] = 0;      bi1[j] = 0;
  }

  const size_t bN = (size_t)b * NN;
  const int rbeg = w * RANGE, rend = rbeg + RANGE;

  // software pipeline: preload chunk 0
  float cx, cy, cz;
  {
    const float* xp = xyz + (bN + rbeg + t) * 3;
    cx = xp[0]; cy = xp[1]; cz = xp[2];
  }

  for (int base = rbeg; base < rend; base += 16) {
    // issue next chunk's load before the compute (branch-free wrap index)
    const int nbase = (base + 16 < rend) ? (base + 16) : rbeg;
    const float* np = xyz + (bN + nbase + t) * 3;
    const float nx = np[0], ny = np[1], nz = np[2];
    if (base + 128 < rend)
      __builtin_prefetch(xyz + (bN + base + 128 + t) * 3, 0, 3);

    const int p = base + t;
    const float pn2 = cx * cx + cy * cy + cz * cz;
    // B operand: lanes 0-15 = K0/K1, lanes 16-31 = K2/K3(=|p|^2)
    v2f bv;
    bv.x = (h == 0) ? cx : cz;
    bv.y = (h == 0) ? cy : pn2;
    v8f acc = __builtin_amdgcn_wmma_f32_16x16x4_f32(
        false, av, false, bv, (short)0, cinit, false, false);

    // d^2 >= 0 -> raw bits are sortable keys (clamp epsilon-negatives to 0)
    unsigned kk[8];
#pragma unroll
    for (int j = 0; j < 8; ++j) kk[j] = f2key(acc[j]);

    // uniform fast-path gate: skip all 8 insert loops if no lane has any
    // qualifying candidate for any of its queries this chunk
    unsigned qual = 0u;
#pragma unroll
    for (int j = 0; j < 8; ++j) qual |= (kk[j] < cmax[j]) ? (1u << j) : 0u;
    if (__any(qual != 0u)) {
#pragma unroll
      for (int j = 0; j < 8; ++j)
        topk_insert<true>(kk[j], p, base,
                          bd0[j], bd1[j], bi0[j], bi1[j], cmax[j], lane, h);
    }
    cx = nx; cy = ny; cz = nz;
  }

  // ---- merge partial sets (waves 1..3 -> wave 0) through LDS
  __shared__ unsigned lds_d[WSPLIT - 1][8][2][16][2];
  __shared__ int      lds_i[WSPLIT - 1][8][2][16][2];
  if (w > 0) {
#pragma unroll
    for (int j = 0; j < 8; ++j) {
      lds_d[w - 1][j][h][t][0] = bd0[j];
      lds_d[w - 1][j][h][t][1] = bd1[j];
      lds_i[w - 1][j][h][t][0] = bi0[j];
      lds_i[w - 1][j][h][t][1] = bi1[j];
    }
  }
  __syncthreads();
  if (w != 0) return;

  for (int src = 0; src < WSPLIT - 1; ++src) {
#pragma unroll
    for (int j = 0; j < 8; ++j) {
      unsigned k0 = lds_d[src][j][h][t][0]; int p0 = lds_i[src][j][h][t][0];
      unsigned k1 = lds_d[src][j][h][t][1]; int p1 = lds_i[src][j][h][t][1];
      topk_insert<false>(k0, p0, -1, bd0[j], bd1[j], bi0[j], bi1[j], cmax[j], lane, h);
      topk_insert<false>(k1, p1, -1, bd0[j], bd1[j], bi0[j], bi1[j], cmax[j], lane, h);
    }
  }

  // ---- ascending-distance extraction (matches top_k(-d2) ordering)
#pragma unroll
  for (int j = 0; j < 8; ++j) {
    const int mq = mtile * 16 + j + 8 * h;
    const size_t rowbase = ((size_t)b * MM + mq) * NS;
    for (int r = 0; r < NS; ++r) {
      const unsigned mymin = (bd0[j] < bd1[j]) ? bd0[j] : bd1[j];
      const unsigned v = row_min16_u32(mymin);
      const bool oc = (mymin == v);
      const unsigned om = ((unsigned)__ballot(oc) >> (h * 16)) & 0xFFFFu;
      const int owner = h * 16 + __builtin_ctz(om | 0x10000u);
      if (lane == owner) {
        if (bd0[j] <= bd1[j]) { idxbuf[rowbase + r] = bi0[j]; bd0[j] = EMPTYK; }
        else                  { idxbuf[rowbase + r] = bi1[j]; bd1[j] = EMPTYK; }
      }
    }
  }
}

// Phase 2: group/gather. One wave per query; lane = sample slot.
// Writes are fully coalesced (32 consecutive floats); feature gathers are
// served from L2 (features = 16.8 MB << 192 MB L2).
__global__ __launch_bounds__(256) void
group_kernel(const float* __restrict__ xyz, const float* __restrict__ new_xyz,
             const float* __restrict__ feats, const int* __restrict__ idx,
             float* __restrict__ out) {
  const int w = threadIdx.x >> 5, lane = threadIdx.x & 31;
  const int gq = blockIdx.x * 8 + w;          // global query 0..B*M-1
  const int b = gq / MM, m = gq % MM;
  const int id = idx[(size_t)gq * NS + lane];

  const float* xp = xyz + ((size_t)b * NN + id) * 3;
  const float px = xp[0], py = xp[1], pz = xp[2];
  const float* qp = new_xyz + ((size_t)b * MM + m) * 3;
  const float qx = qp[0], qy = qp[1], qz = qp[2];

  float* NF = out;                                    // (B, 67, M, 32)
  float* GX = out + (size_t)BB * CH * MM * NS;        // (B, 3, M, 32)
  const size_t stride = (size_t)MM * NS;

  const size_t gxb = ((size_t)b * 3 * MM + m) * NS + lane;
  GX[gxb]              = px;
  GX[gxb + stride]     = py;
  GX[gxb + 2 * stride] = pz;

  const size_t nfb = ((size_t)b * CH * MM + m) * NS + lane;
  NF[nfb]              = px - qx;
  NF[nfb + stride]     = py - qy;
  NF[nfb + 2 * stride] = pz - qz;

  const float* fb = feats + (size_t)b * CC * NN + id;
  float* ob = NF + nfb + 3 * stride;
#pragma unroll 8
  for (int ch = 0; ch < CC; ++ch)
    ob[(size_t)ch * stride] = fb[(size_t)ch * NN];
}

extern "C" void kernel_launch(void* const* d_in, const int* in_sizes, int n_in,
                              void* d_out, int out_size, void* d_ws, size_t ws_size,
                              hipStream_t stream) {
  const float* xyz     = (const float*)d_in[0];   // (4,16384,3)
  const float* new_xyz = (const float*)d_in[1];   // (4,4096,3)
  const float* feats   = (const float*)d_in[2];   // (4,64,16384)
  float* out = (float*)d_out;
  int* idxbuf = (int*)d_ws;                       // (4,4096,32) int32 = 2 MB

  knn_kernel<<<dim3(BB * (MM / 16)), WSPLIT * 32, 0, stream>>>(xyz, new_xyz, idxbuf);
  group_kernel<<<dim3(BB * MM / 8), 256, 0, stream>>>(xyz, new_xyz, feats, idxbuf, out);
}